// LstmEventSpecific_7799660610147
// MI455X (gfx1250) — compile-verified
//
#include <hip/hip_runtime.h>

typedef __attribute__((ext_vector_type(2))) float v2f;
typedef __attribute__((ext_vector_type(8))) float v8f;

#define Bn 4096
#define Tn 512
#define Dn 32
#define Hn 12
#define Gn 48

__device__ __forceinline__ float sigmoidf_(float x) {
    return 1.0f / (1.0f + __expf(-x));
}
__device__ __forceinline__ float tanhf_(float x) {
    float e = __expf(2.0f * x);
    return (e - 1.0f) / (e + 1.0f);
}

// One wave (32 lanes) handles a 16-row batch tile for all T timesteps.
// gates[16x48] = bias + x_t[16x32] @ Wih^T + h[16x12] @ Whh^T  via fp32 WMMA 16x16x4.
// x loads are software-pipelined one timestep ahead (x_{t+1} is h-independent).
__global__ __launch_bounds__(32) void lstm_fused_wmma_kernel(
    const float* __restrict__ x,    // [B, T, 32]
    const float* __restrict__ Wih,  // [48, 32]
    const float* __restrict__ Whh,  // [48, 12]
    const float* __restrict__ bih,  // [48]
    const float* __restrict__ bhh,  // [48]
    float* __restrict__ out)        // [B*T*12] ++ [B*12] ++ [B*12]
{
    __shared__ float ls_gates[16 * Gn];  // 3 KB
    __shared__ float ls_h[16 * Hn];      // 768 B

    const int lane = threadIdx.x & 31;
    const int half = lane >> 4;   // 0: lanes 0-15, 1: lanes 16-31
    const int lm   = lane & 15;
    const int b0   = blockIdx.x * 16;

    // ---- Preload B-matrix fragments (K = kc*4 + 2*half + v, N = nt*16 + lm) ----
    // For C = A @ B with B[d][g] = W[g][d]: lane holds W[(n)*K + k] as contiguous float2.
    v2f wix[3][8];
    #pragma unroll
    for (int nt = 0; nt < 3; ++nt)
        #pragma unroll
        for (int kc = 0; kc < 8; ++kc)
            wix[nt][kc] = *(const v2f*)(Wih + (nt * 16 + lm) * Dn + kc * 4 + 2 * half);

    v2f whf[3][3];
    #pragma unroll
    for (int nt = 0; nt < 3; ++nt)
        #pragma unroll
        for (int kc = 0; kc < 3; ++kc)
            whf[nt][kc] = *(const v2f*)(Whh + (nt * 16 + lm) * Hn + kc * 4 + 2 * half);

    float bias[3];
    #pragma unroll
    for (int nt = 0; nt < 3; ++nt)
        bias[nt] = bih[nt * 16 + lm] + bhh[nt * 16 + lm];

    // ---- Init state: h = 0 (LDS), c = 0 (registers, 6 elems/lane, 32*6 == 16*12) ----
    for (int i = lane; i < 16 * Hn; i += 32) ls_h[i] = 0.0f;
    float creg[6] = {0.f, 0.f, 0.f, 0.f, 0.f, 0.f};
    __syncthreads();

    // A-fragment base for this lane's x row (row lm, K offset 2*half).
    const float* xbase = x + (size_t)(b0 + lm) * Tn * Dn + 2 * half;

    // Prime the pipeline: load x fragments for t = 0.
    v2f ax[8];
    #pragma unroll
    for (int kc = 0; kc < 8; ++kc) ax[kc] = *(const v2f*)(xbase + kc * 4);

    for (int t = 0; t < Tn; ++t) {
        // ---- Issue next-step x loads first (overlaps with whole step body).
        // Clamped index keeps the last iteration in-bounds (redundant reload of t).
        const int tn = (t + 1 < Tn) ? (t + 1) : t;
        const float* xr = xbase + (size_t)tn * Dn;
        v2f axn[8];
        #pragma unroll
        for (int kc = 0; kc < 8; ++kc) axn[kc] = *(const v2f*)(xr + kc * 4);

        // A-fragments from h (LDS): row lm, K = kc*4 + 2*half + {0,1}
        v2f ah[3];
        #pragma unroll
        for (int kc = 0; kc < 3; ++kc) {
            int off = lm * Hn + kc * 4 + 2 * half;
            ah[kc].x = ls_h[off];
            ah[kc].y = ls_h[off + 1];
        }

        // gates = bias + x@Wih^T + h@Whh^T : 33 v_wmma_f32_16x16x4_f32 per step
        #pragma unroll
        for (int nt = 0; nt < 3; ++nt) {
            float bv = bias[nt];
            v8f acc = {bv, bv, bv, bv, bv, bv, bv, bv};
            #pragma unroll
            for (int kc = 0; kc < 8; ++kc)
                acc = __builtin_amdgcn_wmma_f32_16x16x4_f32(
                    false, ax[kc], false, wix[nt][kc], (short)0, acc, false, false);
            #pragma unroll
            for (int kc = 0; kc < 3; ++kc)
                acc = __builtin_amdgcn_wmma_f32_16x16x4_f32(
                    false, ah[kc], false, whf[nt][kc], (short)0, acc, false, false);
            // C layout: VGPR j -> row (j + 8*half), col lm
            #pragma unroll
            for (int j = 0; j < 8; ++j)
                ls_gates[(j + 8 * half) * Gn + nt * 16 + lm] = acc[j];
        }
        __syncthreads();  // single-wave WG: barrier ~NOP, forces s_wait_dscnt

        // Elementwise gate math: lane owns elements e = lane*6 .. lane*6+5
        #pragma unroll
        for (int i = 0; i < 6; ++i) {
            int e  = lane * 6 + i;
            int m  = e / 12;
            int hh = e % 12;
            const float* gr = ls_gates + m * Gn + hh;
            float ig = sigmoidf_(gr[0]);
            float fg = sigmoidf_(gr[12]);
            float gg = tanhf_(gr[24]);
            float og = sigmoidf_(gr[36]);
            float c  = fg * creg[i] + ig * gg;
            creg[i]  = c;
            float h  = og * tanhf_(c);
            ls_h[m * Hn + hh] = h;
            out[((size_t)(b0 + m) * Tn + t) * Hn + hh] = h;
        }
        __syncthreads();

        // ---- Rotate the software pipeline.
        #pragma unroll
        for (int kc = 0; kc < 8; ++kc) ax[kc] = axn[kc];
    }

    // ---- Final hn / cn ----
    const size_t oh = (size_t)Bn * Tn * Hn;
    const size_t oc = oh + (size_t)Bn * Hn;
    #pragma unroll
    for (int i = 0; i < 6; ++i) {
        int e  = lane * 6 + i;
        int m  = e / 12;
        int hh = e % 12;
        out[oh + (size_t)(b0 + m) * Hn + hh] = ls_h[m * Hn + hh];
        out[oc + (size_t)(b0 + m) * Hn + hh] = creg[i];
    }
}

extern "C" void kernel_launch(void* const* d_in, const int* in_sizes, int n_in,
                              void* d_out, int out_size, void* d_ws, size_t ws_size,
                              hipStream_t stream) {
    const float* x   = (const float*)d_in[0];
    const float* Wih = (const float*)d_in[1];
    const float* Whh = (const float*)d_in[2];
    const float* bih = (const float*)d_in[3];
    const float* bhh = (const float*)d_in[4];
    float* out = (float*)d_out;

    lstm_fused_wmma_kernel<<<Bn / 16, 32, 0, stream>>>(x, Wih, Whh, bih, bhh, out);
}